// GNN_25305947308618
// MI455X (gfx1250) — compile-verified
//
#include <hip/hip_runtime.h>
#include <math.h>

typedef __attribute__((ext_vector_type(2))) float v2f;
typedef __attribute__((ext_vector_type(8))) float v8f;

#define NN 50000
#define EE 625000
#define HH 128
#define GG 1000
#define NF 9
#define AV 119

// ---------------- Atom encoder: h0[n,j] = sum_f emb[f, x[n,f], j] ----------------
__global__ void gnn_atom_enc(const int* __restrict__ x, const float* __restrict__ emb,
                             float* __restrict__ h0) {
    int n = blockIdx.x;
    int j = threadIdx.x;
    float acc = 0.f;
#pragma unroll
    for (int f = 0; f < NF; ++f) {
        int idx = x[n * NF + f];
        acc += emb[((size_t)f * AV + idx) * HH + j];
    }
    h0[(size_t)n * HH + j] = acc;
}

// ---------------- Degree / dinv ----------------
__global__ void gnn_deg_init(float* __restrict__ deg) {
    int n = blockIdx.x * blockDim.x + threadIdx.x;
    if (n < NN) deg[n] = 1.0f;  // self loop
}
__global__ void gnn_deg_count(const int* __restrict__ col, float* __restrict__ deg) {
    int e = blockIdx.x * blockDim.x + threadIdx.x;
    if (e < EE) atomicAdd(&deg[col[e]], 1.0f);
}
__global__ void gnn_deg_rsqrt(float* __restrict__ deg) {
    int n = blockIdx.x * blockDim.x + threadIdx.x;
    if (n < NN) deg[n] = rsqrtf(deg[n]);  // deg >= 1 always
}

// ---------------- WMMA GEMM: hw = (RELU? max(h,0):h) @ W   [N,128]x[128,128] ----------------
// Block: 256 threads = 8 waves; block b -> rows [16b,16b+16), wave w -> cols [16w,16w+16)
// V_WMMA_F32_16X16X4_F32 layouts (wave32):
//   A 16x4 : lane L holds M=L&15 ; VGPR j holds K = j + (L>>4)*2
//   B 4x16 : lane L holds N=L&15 ; VGPR j holds K = j + (L>>4)*2
//   C 16x16: lane L holds N=L&15 ; VGPR j holds M = j + (L>>4)*8
template <bool RELU>
__global__ __launch_bounds__(256) void gnn_gemm_wmma(const float* __restrict__ h,
                                                     const float* __restrict__ W,
                                                     float* __restrict__ hw) {
    __shared__ float Ws[HH * HH];  // 64 KB
    {
        const float4* Wv = (const float4*)W;
        float4* Sv = (float4*)Ws;
        for (int i = threadIdx.x; i < HH * HH / 4; i += 256) Sv[i] = Wv[i];
    }
    __syncthreads();

    const int lane = threadIdx.x & 31;
    const int wave = threadIdx.x >> 5;       // 0..7 -> col tile
    const int n0 = wave * 16;
    const int m0 = blockIdx.x * 16;
    const int mlo = lane & 15;
    const int kk = (lane >> 4) << 1;         // 0 or 2

    const float* hrow = h + (size_t)(m0 + mlo) * HH;
    const float* bcol = Ws + n0 + mlo;

    v8f c = {};
#pragma unroll
    for (int k0 = 0; k0 < HH; k0 += 4) {
        float a0 = hrow[k0 + kk];
        float a1 = hrow[k0 + kk + 1];
        if (RELU) { a0 = fmaxf(a0, 0.f); a1 = fmaxf(a1, 0.f); }
        v2f a; a.x = a0; a.y = a1;
        v2f b; b.x = bcol[(k0 + kk) * HH]; b.y = bcol[(k0 + kk + 1) * HH];
        c = __builtin_amdgcn_wmma_f32_16x16x4_f32(false, a, false, b,
                                                  (short)0, c, false, false);
    }

    const int nc = n0 + mlo;
    const int mbase = m0 + ((lane >> 4) << 3);
#pragma unroll
    for (int j = 0; j < 8; ++j) hw[(size_t)(mbase + j) * HH + nc] = c[j];
}

// ---------------- out[n,:] = hw[n,:] * dinv[n]^2 + bias  (self-loop term) ----------------
__global__ void gnn_self_bias(const float* __restrict__ hw, const float* __restrict__ dinv,
                              const float* __restrict__ bias, float* __restrict__ out) {
    int n = blockIdx.x;
    int j = threadIdx.x;
    float d = dinv[n];
    out[(size_t)n * HH + j] = hw[(size_t)n * HH + j] * d * d + bias[j];
}

// ---------------- edge scatter: out[c,:] += dinv[r]*dinv[c] * hw[r,:] ----------------
__global__ void gnn_scatter(const int* __restrict__ row, const int* __restrict__ col,
                            const float* __restrict__ dinv, const float* __restrict__ hw,
                            float* __restrict__ out) {
    int t = blockIdx.x * blockDim.x + threadIdx.x;  // E*32 items, float4 each
    if (t >= EE * 32) return;
    int e = t >> 5, q = t & 31;
    int r = row[e], c = col[e];
    float nrm = dinv[r] * dinv[c];
    float4 v = ((const float4*)hw)[(size_t)r * 32 + q];
    float* o = out + (size_t)c * HH + q * 4;
    atomicAdd(o + 0, nrm * v.x);
    atomicAdd(o + 1, nrm * v.y);
    atomicAdd(o + 2, nrm * v.z);
    atomicAdd(o + 3, nrm * v.w);
}

// ---------------- pooling ----------------
__global__ void gnn_pool_zero(float* __restrict__ sums, float* __restrict__ cnts) {
    int i = blockIdx.x * blockDim.x + threadIdx.x;
    if (i < GG * HH) sums[i] = 0.f;
    if (i < GG) cnts[i] = 0.f;
}
__global__ void gnn_pool_acc(const float* __restrict__ h, const int* __restrict__ batch,
                             float* __restrict__ sums, float* __restrict__ cnts) {
    int n = blockIdx.x;
    int j = threadIdx.x;
    int g = batch[n];
    atomicAdd(&sums[(size_t)g * HH + j], h[(size_t)n * HH + j]);
    if (j == 0) atomicAdd(&cnts[g], 1.0f);
}
__global__ void gnn_head(const float* __restrict__ sums, const float* __restrict__ cnts,
                         const float* __restrict__ lin_w, const float* __restrict__ lin_b,
                         float* __restrict__ out) {
    __shared__ float red[HH];
    int g = blockIdx.x;
    int j = threadIdx.x;
    float inv = 1.0f / fmaxf(cnts[g], 1.0f);
    red[j] = sums[(size_t)g * HH + j] * inv * lin_w[j];
    __syncthreads();
#pragma unroll
    for (int s = 64; s > 0; s >>= 1) {
        if (j < s) red[j] += red[j + s];
        __syncthreads();
    }
    if (j == 0) out[g] = 1.0f / (1.0f + expf(-(red[0] + lin_b[0])));
}

extern "C" void kernel_launch(void* const* d_in, const int* in_sizes, int n_in,
                              void* d_out, int out_size, void* d_ws, size_t ws_size,
                              hipStream_t stream) {
    const int*   x     = (const int*)d_in[0];
    const int*   ei    = (const int*)d_in[1];
    const int*   batch = (const int*)d_in[2];
    const float* emb   = (const float*)d_in[3];
    const float* W1    = (const float*)d_in[4];
    const float* b1    = (const float*)d_in[5];
    const float* W2    = (const float*)d_in[6];
    const float* b2    = (const float*)d_in[7];
    const float* W3    = (const float*)d_in[8];
    const float* b3    = (const float*)d_in[9];
    const float* lin_w = (const float*)d_in[10];
    const float* lin_b = (const float*)d_in[11];
    float* out = (float*)d_out;

    const int* row = ei;
    const int* col = ei + EE;

    // workspace layout
    char* ws = (char*)d_ws;
    float* bufA = (float*)ws;                                   // N*H
    float* bufB = (float*)(ws + (size_t)NN * HH * 4);           // N*H
    float* dinv = (float*)(ws + (size_t)2 * NN * HH * 4);       // N
    float* sums = dinv + ((NN + 63) & ~63);                     // G*H
    float* cnts = sums + GG * HH;                               // G

    // 1. atom encoder -> bufA
    gnn_atom_enc<<<NN, HH, 0, stream>>>(x, emb, bufA);

    // 2. degrees -> dinv
    gnn_deg_init<<<(NN + 255) / 256, 256, 0, stream>>>(dinv);
    gnn_deg_count<<<(EE + 255) / 256, 256, 0, stream>>>(col, dinv);
    gnn_deg_rsqrt<<<(NN + 255) / 256, 256, 0, stream>>>(dinv);

    const float* Wl[3] = {W1, W2, W3};
    const float* bl[3] = {b1, b2, b3};
    for (int l = 0; l < 3; ++l) {
        // hw = (relu? relu(h):h) @ W  : bufA -> bufB
        // ReLU of the previous layer is fused into this GEMM's A-load (l>0)
        if (l == 0)
            gnn_gemm_wmma<false><<<NN / 16, 256, 0, stream>>>(bufA, Wl[l], bufB);
        else
            gnn_gemm_wmma<true><<<NN / 16, 256, 0, stream>>>(bufA, Wl[l], bufB);
        // out = self-loop + bias : bufB -> bufA
        gnn_self_bias<<<NN, HH, 0, stream>>>(bufB, dinv, bl[l], bufA);
        // out += scatter edges
        gnn_scatter<<<(EE * 32 + 255) / 256, 256, 0, stream>>>(row, col, dinv, bufB, bufA);
    }

    // 3. pool + head
    gnn_pool_zero<<<(GG * HH + 255) / 256, 256, 0, stream>>>(sums, cnts);
    gnn_pool_acc<<<NN, HH, 0, stream>>>(bufA, batch, sums, cnts);
    gnn_head<<<GG, HH, 0, stream>>>(sums, cnts, lin_w, lin_b, out);
}